// PermutohedralRefinedUNetX_67705864454847
// MI455X (gfx1250) — compile-verified
//
#include <hip/hip_runtime.h>

#define HH 512
#define WW 512
#define NPIX (HH * WW)

#define ALPHA_B (32.0f / 33.0f)   /* 1/(1+2^-5) */
#define ALPHA_S 0.8f              /* 1/(1+2^-2) */
#define AB_COEF (10.0f * ALPHA_B) /* BI_COMPAT * alpha_b */
#define AS_COEF (3.0f  * ALPHA_S) /* SP_COMPAT * alpha_s */

typedef float v2f __attribute__((ext_vector_type(2)));
typedef float v8f __attribute__((ext_vector_type(8)));

__device__ __forceinline__ float xlane(float v, int srcLane) {
    // wave32 cross-lane gather: lane gets value from srcLane (byte index)
    return __int_as_float(__builtin_amdgcn_ds_bpermute(srcLane << 2, __float_as_int(v)));
}

// -------------------------------------------------------------------------
// zero table A rows [0..rows) and row 0 of table B (row 0 = "missing
// neighbor" slot, must stay 0; B rows 1..M are fully overwritten by pass 0)
__global__ __launch_bounds__(256) void zero_tables(float4* __restrict__ A,
                                                   float4* __restrict__ B,
                                                   int rows) {
    int i = blockIdx.x * 256 + threadIdx.x;
    if (i < rows) {
        A[i] = make_float4(0.f, 0.f, 0.f, 0.f);
        if (i == 0) B[0] = make_float4(0.f, 0.f, 0.f, 0.f);
    }
}

// -------------------------------------------------------------------------
// splat Q (C=4) into table rows (1-based os), 4 f32 atomics per vertex
template <int D1>
__global__ __launch_bounds__(256) void splat_q(const float4* __restrict__ Q,
                                               const int* __restrict__ os,
                                               const float* __restrict__ ws,
                                               float* __restrict__ tab) {
    int p = blockIdx.x * 256 + threadIdx.x;
    if (p >= NPIX) return;
    float4 q = Q[p];
#pragma unroll
    for (int r = 0; r < D1; ++r) {
        int o = os[p * D1 + r];
        float w = ws[p * D1 + r];
        float* row = tab + (size_t)o * 4;
        atomicAdd(row + 0, w * q.x);
        atomicAdd(row + 1, w * q.y);
        atomicAdd(row + 2, w * q.z);
        atomicAdd(row + 3, w * q.w);
    }
}

// splat ones (C=1 in channel .x) for the normalization filter
template <int D1>
__global__ __launch_bounds__(256) void splat_ones(const int* __restrict__ os,
                                                  const float* __restrict__ ws,
                                                  float* __restrict__ tab) {
    int p = blockIdx.x * 256 + threadIdx.x;
    if (p >= NPIX) return;
#pragma unroll
    for (int r = 0; r < D1; ++r) {
        int o = os[p * D1 + r];
        float w = ws[p * D1 + r];
        atomicAdd(tab + (size_t)o * 4, w);
    }
}

// -------------------------------------------------------------------------
// one blur pass, functional (reads src, writes dst): dst[i+1] = src[i+1] +
// 0.5*(src[n0]+src[n1]); n* == 0 hits the always-zero row 0.
__global__ __launch_bounds__(256) void blur_pass(const float4* __restrict__ src,
                                                 float4* __restrict__ dst,
                                                 const int* __restrict__ bnj,
                                                 int M) {
    int i = blockIdx.x * 256 + threadIdx.x;
    if (i >= M) return;
    int n0 = bnj[2 * i + 0];
    int n1 = bnj[2 * i + 1];
    float4 a = src[i + 1];
    float4 x = src[n0];
    float4 y = src[n1];
    dst[i + 1] = make_float4(a.x + 0.5f * (x.x + y.x),
                             a.y + 0.5f * (x.y + y.y),
                             a.z + 0.5f * (x.z + y.z),
                             a.w + 0.5f * (x.w + y.w));
}

// -------------------------------------------------------------------------
// slice the ones-filter and form inv normalization: 1/(alpha*slice + 1e-20)
template <int D1>
__global__ __launch_bounds__(256) void slice_norm(const float* __restrict__ tab,
                                                  const int* __restrict__ os,
                                                  const float* __restrict__ ws,
                                                  float* __restrict__ inv,
                                                  float alpha) {
    int p = blockIdx.x * 256 + threadIdx.x;
    if (p >= NPIX) return;
    float s = 0.f;
#pragma unroll
    for (int r = 0; r < D1; ++r) {
        int o = os[p * D1 + r];
        float w = ws[p * D1 + r];
        s += w * tab[(size_t)o * 4];  // channel .x of float4 row
    }
    inv[p] = 1.0f / (alpha * s + 1e-20f);
}

// -------------------------------------------------------------------------
// Q0 = softmax(-unary)
__global__ __launch_bounds__(256) void init_q(const float* __restrict__ u,
                                              float4* __restrict__ q) {
    int p = blockIdx.x * 256 + threadIdx.x;
    if (p >= NPIX) return;
    float4 uu = ((const float4*)u)[p];
    float l0 = -uu.x, l1 = -uu.y, l2 = -uu.z, l3 = -uu.w;
    float mx = fmaxf(fmaxf(l0, l1), fmaxf(l2, l3));
    float e0 = __expf(l0 - mx), e1 = __expf(l1 - mx);
    float e2 = __expf(l2 - mx), e3 = __expf(l3 - mx);
    float r = 1.0f / (e0 + e1 + e2 + e3);
    q[p] = make_float4(e0 * r, e1 * r, e2 * r, e3 * r);
}

// -------------------------------------------------------------------------
// Fused slice(bilateral+spatial) + message + softmax update.
// 2 lanes per pixel: lane l and lane l^16 split the 9 vertex gathers and own
// channels {0,1} / {2,3}. The exp values then sit exactly in the documented
// 16x4 f32 WMMA A-matrix layout (lanes 0-15: K=0,1; lanes 16-31: K=2,3), so
// the softmax denominator (row-sum over channels for 16 pixels) is computed
// with V_WMMA_F32_16X16X4_F32 against an all-ones B (layout-invariant).
__global__ __launch_bounds__(256) void combine_wmma(
    const float* __restrict__ unary,
    const float4* __restrict__ Tb, const int* __restrict__ osb,
    const float* __restrict__ wsb, const float* __restrict__ invb,
    const float4* __restrict__ Ts, const int* __restrict__ oss,
    const float* __restrict__ wss, const float* __restrict__ invs,
    float* __restrict__ qout) {
    const int tid  = blockIdx.x * 256 + threadIdx.x;
    const int lane = threadIdx.x & 31;
    const int half = lane >> 4;   // 0: channels 0,1 ; 1: channels 2,3
    const int m    = lane & 15;   // pixel slot within wave
    const int p    = (tid >> 5) * 16 + m;  // wave handles 16 pixels

    // bilateral slice: half 0 takes r=0..2, half 1 takes r=3..5
    float4 ab = make_float4(0.f, 0.f, 0.f, 0.f);
#pragma unroll
    for (int j = 0; j < 3; ++j) {
        int r = half * 3 + j;
        int o = osb[p * 6 + r];
        float w = wsb[p * 6 + r];
        float4 t = Tb[o];
        ab.x += w * t.x; ab.y += w * t.y; ab.z += w * t.z; ab.w += w * t.w;
    }
    // spatial slice: half 0 takes r=0,1 ; half 1 takes r=2
    float4 as = make_float4(0.f, 0.f, 0.f, 0.f);
    {
        int r = half ? 2 : 0;
        int o = oss[p * 3 + r];
        float w = wss[p * 3 + r];
        float4 t = Ts[o];
        as.x += w * t.x; as.y += w * t.y; as.z += w * t.z; as.w += w * t.w;
    }
    if (!half) {
        int o = oss[p * 3 + 1];
        float w = wss[p * 3 + 1];
        float4 t = Ts[o];
        as.x += w * t.x; as.y += w * t.y; as.z += w * t.z; as.w += w * t.w;
    }

    float cb = AB_COEF * invb[p];
    float cs = AS_COEF * invs[p];
    float pmx = cb * ab.x + cs * as.x;
    float pmy = cb * ab.y + cs * as.y;
    float pmz = cb * ab.z + cs * as.z;
    float pmw = cb * ab.w + cs * as.w;

    // sum partner's partial slice -> full 4-channel message
    const int partner = lane ^ 16;
    float mgx = pmx + xlane(pmx, partner);
    float mgy = pmy + xlane(pmy, partner);
    float mgz = pmz + xlane(pmz, partner);
    float mgw = pmw + xlane(pmw, partner);

    // logits for this lane's two channels: -u + msg (COMPATIBILITY = -1)
    const int c0 = half * 2;
    float m0 = half ? mgz : mgx;
    float m1 = half ? mgw : mgy;
    float2 uu = *(const float2*)(unary + (size_t)p * 4 + c0);
    float l0 = -uu.x + m0;
    float l1 = -uu.y + m1;

    // pairwise max for stable softmax
    float tm = fmaxf(l0, l1);
    float mx = fmaxf(tm, xlane(tm, partner));
    float e0 = __expf(l0 - mx);
    float e1 = __expf(l1 - mx);

    // --- WMMA denominator: A[16x4] = exp values (native layout), B = ones
    v2f a;  a[0] = e0;   a[1] = e1;
    v2f b;  b[0] = 1.0f; b[1] = 1.0f;
    v8f c = {};
    c = __builtin_amdgcn_wmma_f32_16x16x4_f32(false, a, false, b,
                                              (short)0, c, false, false);
    // C/D layout: VGPR v, lanes 0-15 -> M=v, lanes 16-31 -> M=v+8.
    // Extract c[lane&7], then route rowsum of pixel slot m to its lanes:
    // rowsum_m lives on lane m (m<8) or lane m+8 (m>=8).
    int k = lane & 7;
    float s01 = (k & 1) ? c[1] : c[0];
    float s23 = (k & 1) ? c[3] : c[2];
    float s45 = (k & 1) ? c[5] : c[4];
    float s67 = (k & 1) ? c[7] : c[6];
    float sA  = (k & 2) ? s23 : s01;
    float sB  = (k & 2) ? s67 : s45;
    float sl  = (k & 4) ? sB : sA;
    int src = (m < 8) ? m : (m + 8);
    float S  = xlane(sl, src);
    float rS = 1.0f / S;

    *(float2*)(qout + (size_t)p * 4 + c0) = make_float2(e0 * rS, e1 * rS);
}

// -------------------------------------------------------------------------
extern "C" void kernel_launch(void* const* d_in, const int* in_sizes, int n_in,
                              void* d_out, int out_size, void* d_ws,
                              size_t ws_size, hipStream_t stream) {
    (void)n_in; (void)out_size; (void)ws_size;

    const float* unary = (const float*)d_in[0];
    const float* ws_b  = (const float*)d_in[1];
    const int*   os_b  = (const int*)d_in[2];
    const int*   bn_b  = (const int*)d_in[3];  // [6, M_b, 2]
    const float* ws_s  = (const float*)d_in[5];
    const int*   os_s  = (const int*)d_in[6];
    const int*   bn_s  = (const int*)d_in[7];  // [3, M_s, 2]

    const int M_b = in_sizes[3] / 12;  // 6 * M * 2
    const int M_s = in_sizes[7] / 6;   // 3 * M * 2

    // scratch carve-out (256B aligned slices)
    char* base = (char*)d_ws;
    size_t off = 0;
    auto take = [&](size_t bytes) -> char* {
        char* r = base + off;
        off += (bytes + 255) & ~(size_t)255;
        return r;
    };
    float4* tAb = (float4*)take((size_t)(M_b + 1) * sizeof(float4));
    float4* tBb = (float4*)take((size_t)(M_b + 1) * sizeof(float4));
    float4* tAs = (float4*)take((size_t)(M_s + 1) * sizeof(float4));
    float4* tBs = (float4*)take((size_t)(M_s + 1) * sizeof(float4));
    float*  inv_nb = (float*)take((size_t)NPIX * sizeof(float));
    float*  inv_ns = (float*)take((size_t)NPIX * sizeof(float));
    float4* Qbuf   = (float4*)take((size_t)NPIX * sizeof(float4));

    const dim3 blk(256);
    const int pixBlocks = NPIX / 256;
    const int zbB = (M_b + 1 + 255) / 256, zbS = (M_s + 1 + 255) / 256;
    const int blB = (M_b + 255) / 256,     blS = (M_s + 255) / 256;
    const int cmbBlocks = (NPIX / 16) * 32 / 256;  // 2 lanes per pixel

    // ---- normalization maps (filter of ones), computed once ----
    zero_tables<<<zbB, blk, 0, stream>>>(tAb, tBb, M_b + 1);
    splat_ones<6><<<pixBlocks, blk, 0, stream>>>(os_b, ws_b, (float*)tAb);
    {
        float4 *s = tAb, *d = tBb;
        for (int j = 0; j < 6; ++j) {
            blur_pass<<<blB, blk, 0, stream>>>(s, d, bn_b + (size_t)j * M_b * 2, M_b);
            float4* t = s; s = d; d = t;
        }
        slice_norm<6><<<pixBlocks, blk, 0, stream>>>((const float*)s, os_b, ws_b,
                                                     inv_nb, ALPHA_B);
    }
    zero_tables<<<zbS, blk, 0, stream>>>(tAs, tBs, M_s + 1);
    splat_ones<3><<<pixBlocks, blk, 0, stream>>>(os_s, ws_s, (float*)tAs);
    {
        float4 *s = tAs, *d = tBs;
        for (int j = 0; j < 3; ++j) {
            blur_pass<<<blS, blk, 0, stream>>>(s, d, bn_s + (size_t)j * M_s * 2, M_s);
            float4* t = s; s = d; d = t;
        }
        slice_norm<3><<<pixBlocks, blk, 0, stream>>>((const float*)s, os_s, ws_s,
                                                     inv_ns, ALPHA_S);
    }

    // ---- Q0 = softmax(-unary) ----
    init_q<<<pixBlocks, blk, 0, stream>>>(unary, Qbuf);

    // ---- 10 mean-field iterations ----
    for (int it = 0; it < 10; ++it) {
        // bilateral filter of Q
        zero_tables<<<zbB, blk, 0, stream>>>(tAb, tBb, M_b + 1);
        splat_q<6><<<pixBlocks, blk, 0, stream>>>(Qbuf, os_b, ws_b, (float*)tAb);
        float4 *sb = tAb, *db = tBb;
        for (int j = 0; j < 6; ++j) {
            blur_pass<<<blB, blk, 0, stream>>>(sb, db, bn_b + (size_t)j * M_b * 2, M_b);
            float4* t = sb; sb = db; db = t;
        }
        // spatial filter of Q
        zero_tables<<<zbS, blk, 0, stream>>>(tAs, tBs, M_s + 1);
        splat_q<3><<<pixBlocks, blk, 0, stream>>>(Qbuf, os_s, ws_s, (float*)tAs);
        float4 *ss = tAs, *ds = tBs;
        for (int j = 0; j < 3; ++j) {
            blur_pass<<<blS, blk, 0, stream>>>(ss, ds, bn_s + (size_t)j * M_s * 2, M_s);
            float4* t = ss; ss = ds; ds = t;
        }
        // fused slice + message + softmax (last iteration writes d_out)
        float* outp = (it == 9) ? (float*)d_out : (float*)Qbuf;
        combine_wmma<<<cmbBlocks, blk, 0, stream>>>(unary, sb, os_b, ws_b, inv_nb,
                                                    ss, os_s, ws_s, inv_ns, outp);
    }
}